// Mixer_87290915324439
// MI455X (gfx1250) — compile-verified
//
#include <hip/hip_runtime.h>
#include <hip/hip_bf16.h>
#include <math.h>

// Shapes (fixed by the reference): B=4, C=32, H=W=64, L=4096, DM=64, DI=128, N=16, R=4, K=4
#define LPIX 4096

typedef __attribute__((ext_vector_type(2))) float v2f;
typedef __attribute__((ext_vector_type(8))) float v8f;
typedef unsigned int u32x4 __attribute__((ext_vector_type(4)));
typedef int i32x4 __attribute__((ext_vector_type(4)));
typedef int i32x8 __attribute__((ext_vector_type(8)));

__device__ __forceinline__ float gelu_f(float x) {
    return 0.5f * x * (1.f + erff(x * 0.70710678118654752f));
}
__device__ __forceinline__ float silu_f(float x) {
    return x / (1.f + __expf(-x));
}
template <int ACT>
__device__ __forceinline__ float act_f(float x) {
    if constexpr (ACT == 1) return gelu_f(x);
    if constexpr (ACT == 2) return silu_f(x);
    return x;
}

__device__ __forceinline__ v8f wmma4(v2f a, v2f b, v8f c) {
    return __builtin_amdgcn_wmma_f32_16x16x4_f32(false, a, false, b, (short)0, c, false, false);
}

// ---------------------------------------------------------------------------
// fp32 WMMA GEMM, compile-time specialized:
//   Out[z][m][n] = act( sum_k W[z%wmod][m][k] * (scale[z][k]*X[z][k][n]) + bias[m] + res[z][m][n] )
// One wave computes a 16(M) x 64(N) strip with V_WMMA_F32_16X16X4_F32; K fully unrolled.
// A frag (16x4 f32): lane l, vgpr v -> A[M = l&15][K = 2*(l>>4) + v]
// B frag (4x16 f32): lane l, vgpr v -> B[K = 2*(l>>4) + v][N = l&15]
// C/D (16x16 f32):   lane l, vgpr v -> C[M = v + 8*(l>>4)][N = l&15]
// ---------------------------------------------------------------------------
template <int KDIM, int ACT, bool HAS_SCALE, bool HAS_RES, bool GUARD_M>
__global__ __launch_bounds__(32) void gemm_t(
    const float* __restrict__ W, int wmod, long wstride,
    const float* __restrict__ X, long xstride,
    const float* __restrict__ bias,
    const float* __restrict__ res, long rstride,
    const float* __restrict__ scale,
    float* __restrict__ Out, long ostride,
    int Mvalid, int Ntot)
{
    const int z = blockIdx.z;
    const float* Wz = W + (long)(z % wmod) * wstride;
    const float* Xz = X + (long)z * xstride;
    float* Oz = Out + (long)z * ostride;

    const int lane  = threadIdx.x & 31;
    const int lm    = lane & 15;
    const int half  = lane >> 4;
    const int mbase = blockIdx.y * 16;
    const int nbase = blockIdx.x * 64;
    const int mrow  = mbase + lm;

    v8f acc0 = {}, acc1 = {}, acc2 = {}, acc3 = {};

    const float* xbase = Xz + nbase + lm;
    const float* wrow  = Wz + (long)mrow * KDIM + half * 2;

#pragma unroll
    for (int k0 = 0; k0 < KDIM; k0 += 4) {
        v2f a;
        if constexpr (GUARD_M) {
            if (mrow < Mvalid) { a.x = wrow[k0]; a.y = wrow[k0 + 1]; }
            else               { a.x = 0.f;      a.y = 0.f; }
        } else {
            a.x = wrow[k0];
            a.y = wrow[k0 + 1];
        }
        const int ka = k0 + half * 2;
        float s0 = 1.f, s1 = 1.f;
        if constexpr (HAS_SCALE) {
            const float* sz = scale + (long)z * 64;
            s0 = sz[ka];
            s1 = sz[ka + 1];
        }
        const float* xr0 = xbase + (long)ka * Ntot;
        const float* xr1 = xr0 + Ntot;
        if (k0 + 4 < KDIM) {
            __builtin_prefetch(xr0 + 4 * (long)Ntot, 0, 0);   // global_prefetch_b8
            __builtin_prefetch(xr1 + 4 * (long)Ntot, 0, 0);
        }
        v2f b0, b1, b2, b3;
        b0.x = xr0[0]  * s0;  b0.y = xr1[0]  * s1;
        b1.x = xr0[16] * s0;  b1.y = xr1[16] * s1;
        b2.x = xr0[32] * s0;  b2.y = xr1[32] * s1;
        b3.x = xr0[48] * s0;  b3.y = xr1[48] * s1;

        acc0 = wmma4(a, b0, acc0);
        acc1 = wmma4(a, b1, acc1);
        acc2 = wmma4(a, b2, acc2);
        acc3 = wmma4(a, b3, acc3);
    }

#pragma unroll
    for (int v = 0; v < 8; v++) {
        const int m = mbase + v + half * 8;
        if constexpr (GUARD_M) { if (m >= Mvalid) continue; }
        const float bv = bias ? bias[m] : 0.f;
        const long rowo = (long)m * Ntot + nbase + lm;
        float o0 = acc0[v] + bv, o1 = acc1[v] + bv, o2 = acc2[v] + bv, o3 = acc3[v] + bv;
        if constexpr (HAS_RES) {
            const float* rp = res + (long)z * rstride + rowo;
            o0 += rp[0]; o1 += rp[16]; o2 += rp[32]; o3 += rp[48];
        }
        Oz[rowo +  0] = act_f<ACT>(o0);
        Oz[rowo + 16] = act_f<ACT>(o1);
        Oz[rowo + 32] = act_f<ACT>(o2);
        Oz[rowo + 48] = act_f<ACT>(o3);
    }
}

// ---------------------------------------------------------------------------
// Tensor Data Mover staging: DMA a (4 rows x 64 cols) f32 tile (row stride = 4096
// elements) from global to LDS. D# per CDNA5 ISA ch.8: group0 = {flags, lds_addr,
// global_addr, type=2}; group1 = {data_size=4B, tensor dims/strides, tile dims}.
// Guarded by __has_builtin; arity differs between clang-22 (5 args) and clang-23 (6).
// ---------------------------------------------------------------------------
__device__ __forceinline__ void tdm_load_tile_4x64(const float* gptr, float* lptr) {
#if defined(__has_builtin) && __has_builtin(__builtin_amdgcn_tensor_load_to_lds)
    const unsigned long long ga = (unsigned long long)gptr;
    const unsigned lds = (unsigned)(unsigned long long)lptr;
    u32x4 g0;
    g0.x = 1u;                                               // count=1, user mode
    g0.y = lds;                                              // lds_addr (bytes)
    g0.z = (unsigned)(ga & 0xffffffffull);                   // global_addr[31:0]
    g0.w = (unsigned)((ga >> 32) & 0x1ffffffull) | (2u << 30); // global_addr[56:32] | type=2
    i32x8 g1;
    g1[0] = (2 << 16);              // data_size = 4 bytes
    g1[1] = (int)(4096u << 16);     // tensor_dim0 = 4096 (bits 79:48)
    g1[2] = (64 << 16);             // tensor_dim1 = 64   (bits 111:80)
    g1[3] = (64 << 16);             // tile_dim0  = 64    (bits 127:112)
    g1[4] = 4;                      // tile_dim1  = 4     (bits 143:128)
    g1[5] = 4096;                   // tensor_dim0_stride (bits 207:160)
    g1[6] = 0;
    g1[7] = 0;
    i32x4 gz = {0, 0, 0, 0};
#if __clang_major__ >= 23
    i32x8 gz8 = {0, 0, 0, 0, 0, 0, 0, 0};
    __builtin_amdgcn_tensor_load_to_lds(g0, g1, gz, gz, gz8, 0);
#else
    __builtin_amdgcn_tensor_load_to_lds(g0, g1, gz, gz, 0);
#endif
#else
    // fallback: cooperative plain loads (one wave)
    const int t = threadIdx.x & 31;
    for (int r = 0; r < 4; r++) {
        lptr[r * 64 + t]      = gptr[(long)r * 4096 + t];
        lptr[r * 64 + 32 + t] = gptr[(long)r * 4096 + 32 + t];
    }
#endif
}

__device__ __forceinline__ void tdm_wait0() {
#if defined(__has_builtin) && __has_builtin(__builtin_amdgcn_s_wait_tensorcnt)
    __builtin_amdgcn_s_wait_tensorcnt(0);
#endif
}
__device__ __forceinline__ void tdm_wait1() {
#if defined(__has_builtin) && __has_builtin(__builtin_amdgcn_s_wait_tensorcnt)
    __builtin_amdgcn_s_wait_tensorcnt(1);
#endif
}

// ---------------------------------------------------------------------------
// TDM-staged WMMA GEMM for the big in-projection: M=256, K=64, N=4096 per batch.
// Double-buffered 4x64 B tiles in LDS via tensor_load_to_lds + s_wait_tensorcnt.
// ---------------------------------------------------------------------------
__global__ __launch_bounds__(32) void gemm_tdm_k(
    const float* __restrict__ W,        // 256 x 64 row-major
    const float* __restrict__ X,        // per batch 64 x 4096
    long xstride,
    float* __restrict__ Out, long ostride,
    int Ntot)
{
    constexpr int KDIM = 64;
    __shared__ float bs[2][4 * 64];

    const int z = blockIdx.z;
    const float* Xz = X + (long)z * xstride;
    float* Oz = Out + (long)z * ostride;

    const int lane  = threadIdx.x & 31;
    const int lm    = lane & 15;
    const int half  = lane >> 4;
    const int mbase = blockIdx.y * 16;
    const int nbase = blockIdx.x * 64;
    const int mrow  = mbase + lm;

    v8f acc0 = {}, acc1 = {}, acc2 = {}, acc3 = {};
    const float* wrow = W + (long)mrow * KDIM + half * 2;

    tdm_load_tile_4x64(Xz + nbase, &bs[0][0]);

#pragma unroll
    for (int i = 0; i < KDIM / 4; i++) {
        const int k0 = i * 4;
        if (i + 1 < KDIM / 4) {
            tdm_load_tile_4x64(Xz + (long)(k0 + 4) * Ntot + nbase, &bs[(i + 1) & 1][0]);
            tdm_wait1();            // tile i complete, tile i+1 in flight
        } else {
            tdm_wait0();
        }
        const float* bb = &bs[i & 1][0];
        const int r0 = half * 2, r1 = half * 2 + 1;

        v2f a;
        a.x = wrow[k0];
        a.y = wrow[k0 + 1];
        v2f b0, b1, b2, b3;
        b0.x = bb[r0 * 64 + lm +  0];  b0.y = bb[r1 * 64 + lm +  0];
        b1.x = bb[r0 * 64 + lm + 16];  b1.y = bb[r1 * 64 + lm + 16];
        b2.x = bb[r0 * 64 + lm + 32];  b2.y = bb[r1 * 64 + lm + 32];
        b3.x = bb[r0 * 64 + lm + 48];  b3.y = bb[r1 * 64 + lm + 48];

        acc0 = wmma4(a, b0, acc0);
        acc1 = wmma4(a, b1, acc1);
        acc2 = wmma4(a, b2, acc2);
        acc3 = wmma4(a, b3, acc3);
    }

#pragma unroll
    for (int v = 0; v < 8; v++) {
        const int m = mbase + v + half * 8;
        const long rowo = (long)m * Ntot + nbase + lm;
        Oz[rowo +  0] = acc0[v];
        Oz[rowo + 16] = acc1[v];
        Oz[rowo + 32] = acc2[v];
        Oz[rowo + 48] = acc3[v];
    }
}

// ---------------------------------------------------------------------------
// Depthwise 3x3 conv (cross-correlation, zero pad = dil, rhs dilation = dil)
// ---------------------------------------------------------------------------
__global__ __launch_bounds__(256) void dwconv_k(
    const float* __restrict__ X, long xbs,
    const float* __restrict__ Wc, const float* __restrict__ bc,
    float* __restrict__ Out, long obs, int dil, int act)
{
    const int l  = blockIdx.x * 256 + threadIdx.x;
    const int ch = blockIdx.y;
    const int b  = blockIdx.z;
    const int h = l >> 6, w = l & 63;
    const float* xp = X + (long)b * xbs + (long)ch * LPIX;
    const float* wp = Wc + ch * 9;
    float s = bc[ch];
#pragma unroll
    for (int r = 0; r < 3; r++) {
        const int hh = h + (r - 1) * dil;
        if ((unsigned)hh >= 64u) continue;
#pragma unroll
        for (int c = 0; c < 3; c++) {
            const int ww = w + (c - 1) * dil;
            if ((unsigned)ww >= 64u) continue;
            s += xp[hh * 64 + ww] * wp[r * 3 + c];
        }
    }
    if (act == 2) s = silu_f(s);
    Out[(long)b * obs + (long)ch * LPIX + l] = s;
}

// ---------------------------------------------------------------------------
// Build the 4 scan directions: xs[b,k,d,l]. T(l) = (l&63)*64 + (l>>6)
// ---------------------------------------------------------------------------
__global__ __launch_bounds__(256) void build_xs_k(const float* __restrict__ xc,
                                                  float* __restrict__ xs)
{
    const int l = blockIdx.x * 256 + threadIdx.x;
    const int d = blockIdx.y;
    const int b = blockIdx.z;
    const float* src = xc + ((long)b * 128 + d) * LPIX;
    const long o = (((long)b * 4) * 128 + d) * LPIX;
    const int Tl = ((l & 63) << 6) | (l >> 6);
    xs[o + l]                   = src[l];
    xs[o + 1L * 128 * LPIX + l] = src[Tl];
    xs[o + 2L * 128 * LPIX + l] = src[4095 - l];
    xs[o + 3L * 128 * LPIX + l] = src[4095 - Tl];
}

// ---------------------------------------------------------------------------
// Selective scan. One lane per (b,k,d,n) state; 16 lanes per channel.
// Delta projection + softplus fused. y reduction over n via shfl_xor.
// ---------------------------------------------------------------------------
__global__ __launch_bounds__(256) void scan_k(
    const float* __restrict__ xs, const float* __restrict__ xdbl,
    const float* __restrict__ dt_w, const float* __restrict__ dt_b,
    const float* __restrict__ A_logs, const float* __restrict__ Ds,
    float* __restrict__ oy)
{
    const int gid = blockIdx.x * 256 + threadIdx.x;
    const int n   = gid & 15;
    const int ch  = gid >> 4;        // (b*4 + k)*128 + d
    const int d   = ch & 127;
    const int bk  = ch >> 7;
    const int k   = bk & 3;
    const int kd  = k * 128 + d;

    const float An = -__expf(A_logs[kd * 16 + n]);
    const float Dv = Ds[kd];
    const float w0 = dt_w[kd * 4 + 0], w1 = dt_w[kd * 4 + 1];
    const float w2 = dt_w[kd * 4 + 2], w3 = dt_w[kd * 4 + 3];
    const float db = dt_b[kd];

    const float* up = xs + (long)ch * LPIX;
    const float* dp = xdbl + (long)bk * 36 * LPIX;
    const float* Bp = dp + (long)(4 + n) * LPIX;
    const float* Cp = dp + (long)(20 + n) * LPIX;
    float* yp = oy + (long)ch * LPIX;

    float h = 0.f;
    auto step = [&](float uv, float Bv, float Cv,
                    float p0, float p1, float p2, float p3, int l) {
        float pre = w0 * p0 + w1 * p1 + w2 * p2 + w3 * p3 + db;
        float dt = (pre > 20.f) ? pre : log1pf(__expf(pre));
        h = __expf(dt * An) * h + dt * Bv * uv;
        float part = h * Cv;
        part += __shfl_xor(part, 1);
        part += __shfl_xor(part, 2);
        part += __shfl_xor(part, 4);
        part += __shfl_xor(part, 8);
        if (n == 0) yp[l] = part + uv * Dv;
    };

    for (int l = 0; l < LPIX; l += 4) {
        const float4 u4 = *(const float4*)(up + l);
        const float4 B4 = *(const float4*)(Bp + l);
        const float4 C4 = *(const float4*)(Cp + l);
        const float4 r0 = *(const float4*)(dp + 0 * LPIX + l);
        const float4 r1 = *(const float4*)(dp + 1 * LPIX + l);
        const float4 r2 = *(const float4*)(dp + 2 * LPIX + l);
        const float4 r3 = *(const float4*)(dp + 3 * LPIX + l);
        step(u4.x, B4.x, C4.x, r0.x, r1.x, r2.x, r3.x, l + 0);
        step(u4.y, B4.y, C4.y, r0.y, r1.y, r2.y, r3.y, l + 1);
        step(u4.z, B4.z, C4.z, r0.z, r1.z, r2.z, r3.z, l + 2);
        step(u4.w, B4.w, C4.w, r0.w, r1.w, r2.w, r3.w, l + 3);
    }
}

// ---------------------------------------------------------------------------
// Merge 4 directions, per-pixel LayerNorm over 128 channels, * silu(z).
// Block (32 pixels, 8 channel rows); tile staged in LDS.
// ---------------------------------------------------------------------------
__global__ __launch_bounds__(256) void merge_ln_k(
    const float* __restrict__ oy, const float* __restrict__ xz,
    const float* __restrict__ lg, const float* __restrict__ lb,
    float* __restrict__ yln)
{
    __shared__ float tile[128][33];
    __shared__ float ps[8][32], pq[8][32];
    __shared__ float mn[32], rs[32];

    const int b  = blockIdx.y;
    const int px = threadIdx.x;   // 0..31
    const int ty = threadIdx.y;   // 0..7
    const int l  = blockIdx.x * 32 + px;
    const int Tl  = ((l & 63) << 6) | (l >> 6);
    const int Fl  = 4095 - l;
    const int FTl = 4095 - Tl;
    const float* base = oy + (long)b * 4 * 128 * LPIX;

    for (int d = ty; d < 128; d += 8) {
        float v = base[(long)(0 * 128 + d) * LPIX + l]
                + base[(long)(2 * 128 + d) * LPIX + Fl]
                + base[(long)(1 * 128 + d) * LPIX + Tl]
                + base[(long)(3 * 128 + d) * LPIX + FTl];
        tile[d][px] = v;
    }
    __syncthreads();

    float s = 0.f, q = 0.f;
    for (int d = ty; d < 128; d += 8) { float v = tile[d][px]; s += v; q += v * v; }
    ps[ty][px] = s; pq[ty][px] = q;
    __syncthreads();
    if (ty == 0) {
        float S = 0.f, Q = 0.f;
        for (int j = 0; j < 8; j++) { S += ps[j][px]; Q += pq[j][px]; }
        float m = S * (1.f / 128.f);
        float var = Q * (1.f / 128.f) - m * m;
        mn[px] = m;
        rs[px] = rsqrtf(var + 1e-5f);
    }
    __syncthreads();

    const float m = mn[px], r = rs[px];
    const float* zb = xz + ((long)b * 256 + 128) * LPIX;
    for (int d = ty; d < 128; d += 8) {
        float v = (tile[d][px] - m) * r * lg[d] + lb[d];
        float z = zb[(long)d * LPIX + l];
        yln[((long)b * 128 + d) * LPIX + l] = v * silu_f(z);
    }
}

// concat(x_local, x_global) then gelu; over B*64*L elements
__global__ __launch_bounds__(256) void cat_gelu_k(const float* __restrict__ xl,
                                                  const float* __restrict__ xg,
                                                  float* __restrict__ out)
{
    const long i = (long)blockIdx.x * 256 + threadIdx.x;
    const int l  = (int)(i & 4095);
    const int ch = (int)((i >> 12) & 63);
    const int b  = (int)(i >> 18);
    float v = (ch < 32) ? xl[((long)b * 32 + ch) * LPIX + l]
                        : xg[((long)b * 32 + (ch - 32)) * LPIX + l];
    out[i] = gelu_f(v);
}

// global average pool: one block per (b,ch)
__global__ __launch_bounds__(256) void pool_k(const float* __restrict__ x,
                                              float* __restrict__ p)
{
    __shared__ float sb[256];
    const long base = (long)blockIdx.x * LPIX;
    float s = 0.f;
    for (int l = threadIdx.x; l < LPIX; l += 256) s += x[base + l];
    sb[threadIdx.x] = s;
    __syncthreads();
    for (int o = 128; o > 0; o >>= 1) {
        if ((int)threadIdx.x < o) sb[threadIdx.x] += sb[threadIdx.x + o];
        __syncthreads();
    }
    if (threadIdx.x == 0) p[blockIdx.x] = sb[0] * (1.f / (float)LPIX);
}

// channel-attention MLP: relu(W1 p + b1) -> sigmoid(W2 . + b2). grid=B, 64 thr.
__global__ __launch_bounds__(64) void attn_k(
    const float* __restrict__ p,
    const float* __restrict__ w1, const float* __restrict__ b1,
    const float* __restrict__ w2, const float* __restrict__ b2,
    float* __restrict__ scale)
{
    __shared__ float pv[64];
    __shared__ float a1[32];
    const int b = blockIdx.x, t = threadIdx.x;
    pv[t] = p[b * 64 + t];
    __syncthreads();
    if (t < 32) {
        float s = b1[t];
        for (int j = 0; j < 64; j++) s += w1[t * 64 + j] * pv[j];
        a1[t] = fmaxf(s, 0.f);
    }
    __syncthreads();
    float s = b2[t];
    for (int j = 0; j < 32; j++) s += w2[t * 32 + j] * a1[j];
    scale[b * 64 + t] = 1.f / (1.f + __expf(-s));
}

// batchnorm stats per channel over (B, H, W); grid = 32
__global__ __launch_bounds__(256) void bnstat_k(const float* __restrict__ y,
                                                float* __restrict__ st)
{
    __shared__ float ss[256], sq[256];
    const int ch = blockIdx.x, t = threadIdx.x;
    float s = 0.f, q = 0.f;
    for (int b = 0; b < 4; b++) {
        const float* yp = y + ((long)b * 32 + ch) * LPIX;
        for (int l = t; l < LPIX; l += 256) { float v = yp[l]; s += v; q += v * v; }
    }
    ss[t] = s; sq[t] = q;
    __syncthreads();
    for (int o = 128; o > 0; o >>= 1) {
        if (t < o) { ss[t] += ss[t + o]; sq[t] += sq[t + o]; }
        __syncthreads();
    }
    if (t == 0) {
        float m = ss[0] * (1.f / 16384.f);
        float var = sq[0] * (1.f / 16384.f) - m * m;
        st[ch] = m;
        st[32 + ch] = rsqrtf(var + 1e-5f);
    }
}

__global__ __launch_bounds__(256) void bnapply_k(
    const float* __restrict__ y, const float* __restrict__ st,
    const float* __restrict__ g, const float* __restrict__ bch,
    float* __restrict__ out)
{
    const long i = (long)blockIdx.x * 256 + threadIdx.x;
    const int ch = (int)((i >> 12) & 31);
    float v = (y[i] - st[ch]) * st[32 + ch] * g[ch] + bch[ch];
    out[i] = fmaxf(v, 0.f);
}

// ---------------------------------------------------------------------------
extern "C" void kernel_launch(void* const* d_in, const int* in_sizes, int n_in,
                              void* d_out, int out_size, void* d_ws, size_t ws_size,
                              hipStream_t stream)
{
    (void)in_sizes; (void)n_in; (void)out_size; (void)ws_size;
    const long L = LPIX;

    const float* x         = (const float*)d_in[0];
    const float* w_init    = (const float*)d_in[1];
    const float* b_init    = (const float*)d_in[2];
    const float* w_dw1     = (const float*)d_in[3];
    const float* b_dw1     = (const float*)d_in[4];
    const float* w_dw2     = (const float*)d_in[5];
    const float* b_dw2     = (const float*)d_in[6];
    const float* w_ginit   = (const float*)d_in[7];
    const float* b_ginit   = (const float*)d_in[8];
    const float* w_gfina   = (const float*)d_in[9];
    const float* b_gfina   = (const float*)d_in[10];
    const float* ss_in_w   = (const float*)d_in[11];
    const float* ss_conv_w = (const float*)d_in[12];
    const float* ss_conv_b = (const float*)d_in[13];
    const float* ss_xproj  = (const float*)d_in[14];
    const float* ss_dt_w   = (const float*)d_in[15];
    const float* ss_dt_b   = (const float*)d_in[16];
    const float* ss_Alogs  = (const float*)d_in[17];
    const float* ss_Ds     = (const float*)d_in[18];
    const float* ss_ln_g   = (const float*)d_in[19];
    const float* ss_ln_b   = (const float*)d_in[20];
    const float* ss_out_w  = (const float*)d_in[21];
    const float* w_ca1     = (const float*)d_in[22];
    const float* b_ca1     = (const float*)d_in[23];
    const float* w_ca2     = (const float*)d_in[24];
    const float* b_ca2     = (const float*)d_in[25];
    const float* w_caconv  = (const float*)d_in[26];
    const float* b_caconv  = (const float*)d_in[27];
    const float* bn_g      = (const float*)d_in[28];
    const float* bn_b      = (const float*)d_in[29];
    float* out = (float*)d_out;

    // workspace bump allocation (floats)
    float* ws = (float*)d_ws;
    long o = 0;
    float* xi   = ws + o; o += 4L * 64 * L;     // (B,64,L)
    float* xloc = ws + o; o += 4L * 32 * L;     // (B,32,L)
    float* g    = ws + o; o += 4L * 64 * L;     // (B,64,L)  g0 residual
    float* xz   = ws + o; o += 4L * 256 * L;    // (B,256,L)
    float* xcb  = ws + o; o += 4L * 128 * L;    // (B,128,L) ss conv out; reused as yln
    float* xs   = ws + o; o += 16L * 128 * L;   // (B,4,128,L)
    float* xdbl = ws + o; o += 16L * 36 * L;    // (B*4,36,L)
    float* oy   = ws + o; o += 16L * 128 * L;   // (B,4,128,L)
    // aliases (xcb dead after build_xs; xs dead after scan)
    float* yln   = xcb;
    float* gsum  = xs;
    float* xglob = gsum + 4L * 64 * L;
    float* xcat  = xglob + 4L * 32 * L;
    float* yout  = xcat + 4L * 64 * L;
    float* pbuf  = yout + 4L * 32 * L;    // 256
    float* ascal = pbuf + 256;            // 256
    float* bnst  = ascal + 256;           // 64

    // 1. xi = w_init (64x32) * x + b_init
    gemm_t<32, 0, false, false, false><<<dim3(64, 4, 4), 32, 0, stream>>>(
        w_init, 1, 0, x, 32L * L, b_init, nullptr, 0, nullptr, xi, 64L * L, 64, LPIX);

    // 2. x_local: dw3x3 dil1 on xi[:, :16], dw3x3 dil2 on xi[:, 16:32]
    dwconv_k<<<dim3(16, 16, 4), 256, 0, stream>>>(xi, 64L * L, w_dw1, b_dw1, xloc, 32L * L, 1, 0);
    dwconv_k<<<dim3(16, 16, 4), 256, 0, stream>>>(xi + 16L * L, 64L * L, w_dw2, b_dw2, xloc + 16L * L, 32L * L, 2, 0);

    // 3. g = gelu(w_ginit (64x32) * xi[:, 32:64] + b)
    gemm_t<32, 1, false, false, false><<<dim3(64, 4, 4), 32, 0, stream>>>(
        w_ginit, 1, 0, xi + 32L * L, 64L * L, b_ginit, nullptr, 0, nullptr, g, 64L * L, 64, LPIX);

    // 4. xz = ss_in_w (256x64) * g   — TDM-staged B panels
    gemm_tdm_k<<<dim3(64, 16, 4), 32, 0, stream>>>(
        ss_in_w, g, 64L * L, xz, 256L * L, LPIX);

    // 5. xc = silu(dw3x3(xz[:, :128]))
    dwconv_k<<<dim3(16, 128, 4), 256, 0, stream>>>(xz, 256L * L, ss_conv_w, ss_conv_b, xcb, 128L * L, 1, 2);

    // 6. xs: 4 scan directions
    build_xs_k<<<dim3(16, 128, 4), 256, 0, stream>>>(xcb, xs);

    // 7. x_dbl[b,k] = xproj_w[k] (36x128) * xs[b,k]   (z = b*4+k)
    gemm_t<128, 0, false, false, true><<<dim3(64, 3, 16), 32, 0, stream>>>(
        ss_xproj, 4, 36L * 128, xs, 128L * L, nullptr, nullptr, 0, nullptr, xdbl, 36L * L, 36, LPIX);

    // 8. selective scan (delta projection fused)
    scan_k<<<dim3(128), 256, 0, stream>>>(xs, xdbl, ss_dt_w, ss_dt_b, ss_Alogs, ss_Ds, oy);

    // 9. merge directions + LN + silu(z) gate
    merge_ln_k<<<dim3(128, 4), dim3(32, 8), 0, stream>>>(oy, xz, ss_ln_g, ss_ln_b, yln);

    // 10. gsum = ss_out_w (64x128) * yln + g0
    gemm_t<128, 0, false, true, false><<<dim3(64, 4, 4), 32, 0, stream>>>(
        ss_out_w, 1, 0, yln, 128L * L, nullptr, g, 64L * L, nullptr, gsum, 64L * L, 64, LPIX);

    // 11. x_global = gelu(w_gfina (32x64) * gsum + b)
    gemm_t<64, 1, false, false, false><<<dim3(64, 2, 4), 32, 0, stream>>>(
        w_gfina, 1, 0, gsum, 64L * L, b_gfina, nullptr, 0, nullptr, xglob, 32L * L, 32, LPIX);

    // 12. xcat = gelu(concat(x_local, x_global))
    cat_gelu_k<<<dim3(4096), 256, 0, stream>>>(xloc, xglob, xcat);

    // 13-14. channel attention
    pool_k<<<dim3(256), 256, 0, stream>>>(xcat, pbuf);
    attn_k<<<dim3(4), 64, 0, stream>>>(pbuf, w_ca1, b_ca1, w_ca2, b_ca2, ascal);

    // 15. yout = w_caconv (32x64) * (ascal ⊙ xcat) + b_caconv  (scale folded into B loads)
    gemm_t<64, 0, true, false, false><<<dim3(64, 2, 4), 32, 0, stream>>>(
        w_caconv, 1, 0, xcat, 64L * L, b_caconv, nullptr, 0, ascal, yout, 32L * L, 32, LPIX);

    // 16-17. batchnorm over (B,H,W) + relu
    bnstat_k<<<dim3(32), 256, 0, stream>>>(yout, bnst);
    bnapply_k<<<dim3(2048), 256, 0, stream>>>(yout, bnst, bn_g, bn_b, out);
}